// BezierSurv_85650237817335
// MI455X (gfx1250) — compile-verified
//
#include <hip/hip_runtime.h>

typedef __attribute__((ext_vector_type(2))) float v2f;
typedef __attribute__((ext_vector_type(4))) float v4f;
typedef __attribute__((ext_vector_type(8))) float v8f;

#define MARGIN   0.1f
#define NCLS     16
#define SBLK     256
#define ROWLEN   4096   // NCLS * SBLK

// Each block: 32 rows. 8 waves: wave w -> row-half (w&1)*16, class-quarter (w>>1)*4.
// Per wave, per class: stream the 256-col class block with b128 loads (lane l
// and l+16 cover adjacent float4s of the same row), accumulate 4 partial sums
// per lane, fold to the 16x4 f32 A-tile layout (2 VGPRs/lane), then one
// V_WMMA_F32_16X16X4_F32 with a one-hot B column reduces K and routes the
// class sum into D[:, class].
__global__ __launch_bounds__(256) void surv_blocksum_kernel(
    const float* __restrict__ sim,
    const int*   __restrict__ label,
    const int*   __restrict__ censor,
    float*       __restrict__ partial)
{
    __shared__ float lds_acc[2][16][16];   // [row-half][m][class]

    const int tid  = threadIdx.x;
    const int lane = tid & 31;
    const int wave = tid >> 5;
    const int q    = wave >> 1;      // class quarter 0..3
    const int h    = wave & 1;       // row half 0..1
    const int n    = lane & 15;      // A-layout: M index / B,D-layout: N index
    const int hi   = lane >> 4;      // A-layout: K pair select

    const int rowTile = blockIdx.x * 32 + h * 16;
    const int row     = rowTile + n;

    const v4f* __restrict__ rp =
        (const v4f*)(sim + (size_t)row * ROWLEN);

    v8f D = {0.f, 0.f, 0.f, 0.f, 0.f, 0.f, 0.f, 0.f};

    for (int cc = 0; cc < 4; ++cc) {
        const int c = q * 4 + cc;
        // class block c = float4 indices [c*64, c*64+64); lane takes the
        // alternating half selected by hi: idx = c*64 + j*2 + hi, j = 0..31
        v4f acc = {0.f, 0.f, 0.f, 0.f};
        const int base = c * 64 + hi;
        #pragma unroll 8
        for (int j = 0; j < 32; ++j) {
            acc += rp[base + j * 2];
        }
        // Fold 4 per-lane partials into the lane's two A K-slots. All B rows
        // are the identical one-hot row e_c, so D row-sums over K regardless
        // of how the partials are split across K slots.
        const v2f a2 = {acc.x + acc.z, acc.y + acc.w};
        const float b = (n == c) ? 1.0f : 0.0f;
        const v2f Bv = {b, b};
        // D[m][nn] += sum_k A[m][k] * (nn == c)
        D = __builtin_amdgcn_wmma_f32_16x16x4_f32(
                /*neg_a=*/false, a2, /*neg_b=*/false, Bv,
                /*c_mod=*/(short)0, D, /*reuse_a=*/false, /*reuse_b=*/false);
    }

    // D layout: lane holds column n = lane&15, rows m = v + 8*hi (v = VGPR idx).
    // Only this wave's 4 class columns are non-zero; quarters are disjoint,
    // so plain stores (no atomics, no init) fill lds_acc completely.
    if (n >= q * 4 && n < q * 4 + 4) {
        #pragma unroll
        for (int v = 0; v < 8; ++v)
            lds_acc[h][v + 8 * hi][n] = D[v];
    }
    __syncthreads();

    // Per-row loss terms: threads 0..31 handle the block's 32 rows.
    if (tid < 32) {
        const int r = blockIdx.x * 32 + tid;
        float cs[NCLS];
        float total = 0.f;
        #pragma unroll
        for (int k = 0; k < NCLS; ++k) {
            cs[k] = lds_acc[tid >> 4][tid & 15][k];
            total += cs[k];
        }
        const int cls = label[r];
        const int cen = censor[r];

        float pos_sum, pos_cnt;
        if (cen == 0) {                    // uncensored: pos = (col_cls == cls)
            pos_sum = cs[cls];
            pos_cnt = (float)SBLK;
        } else if (cls == 0) {             // censored, cls==0: pos = everything
            pos_sum = total;
            pos_cnt = (float)ROWLEN;
        } else {                           // censored: pos = (col_cls >= cls)
            pos_sum = 0.f;
            for (int k = cls; k < NCLS; ++k) pos_sum += cs[k];
            pos_cnt = (float)((NCLS - cls) * SBLK);
        }
        const float neg_cnt  = (float)ROWLEN - pos_cnt;
        const float neg_sum  = total - pos_sum;
        const float pos_mean = pos_sum / pos_cnt;               // pos_cnt >= 256 always
        const float neg_mean = (neg_cnt > 0.f) ? (neg_sum / neg_cnt) : 0.f;
        float term = fmaxf(MARGIN - pos_mean + neg_mean, 0.f);

        // reduce 32 terms within wave 0
        #pragma unroll
        for (int off = 16; off > 0; off >>= 1)
            term += __shfl_down(term, off, 32);
        if (tid == 0)
            partial[blockIdx.x] = term;
    }
}

__global__ __launch_bounds__(256) void surv_reduce_kernel(
    const float* __restrict__ partial, float* __restrict__ out,
    int nPartial, float invB)
{
    __shared__ float s[256];
    float x = 0.f;
    for (int i = threadIdx.x; i < nPartial; i += 256)
        x += partial[i];
    s[threadIdx.x] = x;
    __syncthreads();
    for (int stride = 128; stride > 0; stride >>= 1) {
        if ((int)threadIdx.x < stride)
            s[threadIdx.x] += s[threadIdx.x + stride];
        __syncthreads();
    }
    if (threadIdx.x == 0)
        out[0] = s[0] * invB;
}

extern "C" void kernel_launch(void* const* d_in, const int* in_sizes, int n_in,
                              void* d_out, int out_size, void* d_ws, size_t ws_size,
                              hipStream_t stream) {
    (void)n_in; (void)out_size; (void)ws_size;
    const float* sim    = (const float*)d_in[0];
    const int*   label  = (const int*)  d_in[1];
    const int*   censor = (const int*)  d_in[2];
    // d_in[3] = sample_times (S=256, baked into the kernel tiling)

    const int B = in_sizes[1];           // 16384 rows
    float* out     = (float*)d_out;
    float* partial = (float*)d_ws;

    const int nBlocks = B / 32;          // 512 blocks x 8 waves = 4096 waves
    surv_blocksum_kernel<<<nBlocks, 256, 0, stream>>>(sim, label, censor, partial);
    surv_reduce_kernel<<<1, 256, 0, stream>>>(partial, out, nBlocks, 1.0f / (float)B);
}